// SurfacenessFrangiEnhance_87969520156916
// MI455X (gfx1250) — compile-verified
//
#include <hip/hip_runtime.h>
#include <cmath>

// Volume geometry (fixed by the reference): B=2, D=192, H=256, W=256, f32.
#define DIMD 192
#define DIMH 256
#define DIMW 256
#define VOLB (DIMD * DIMH * DIMW)       // 12,582,912 voxels per batch element
#define NTOT (2 * VOLB)                 // 25,165,824 total
#define EPSF 1e-7f

typedef __attribute__((ext_vector_type(8))) float v8f;
typedef __attribute__((ext_vector_type(2))) float v2f;

__device__ __forceinline__ int iclamp(int v, int lo, int hi) {
  return v < lo ? lo : (v > hi ? hi : v);
}

// ---------------------------------------------------------------------------
// init: zero the 4 per-(batch,sigma) max slots (bit pattern 0 == 0.0f)
// ---------------------------------------------------------------------------
__global__ __launch_bounds__(32) void init_max_kernel(unsigned* mx) {
  if (threadIdx.x < 4) mx[threadIdx.x] = 0u;
}

// ---------------------------------------------------------------------------
// Pass 1: threshold + Gaussian blur along x. Branch-free: clamp the index,
// load unconditionally, select 0 outside (zero 'SAME' padding).
// ---------------------------------------------------------------------------
template <int K> struct WgtArr { float w[K]; };

template <int K, int R>
__global__ __launch_bounds__(256)
void blur_x_thresh_kernel(const float* __restrict__ src, float* __restrict__ dst,
                          WgtArr<K> wg) {
  const int idx = blockIdx.x * 256 + threadIdx.x;
  const int x = idx & (DIMW - 1);
  const int rowBase = idx - x;
  float acc = 0.f;
#pragma unroll
  for (int k = 0; k < K; ++k) {
    const int xx = x + k - R;
    const int xc = iclamp(xx, 0, DIMW - 1);
    const float raw = src[rowBase + xc];
    const float val = ((unsigned)xx < (unsigned)DIMW && raw >= 0.75f) ? 1.f : 0.f;
    acc = fmaf(wg.w[k], val, acc);
  }
  dst[idx] = acc;
}

// ---------------------------------------------------------------------------
// Pass 2/3: Gaussian blur along a strided axis (y or z) via WMMA.
//   D(16x16) = sum_c A_c(16x4) * B_c(4x16) + C  with
//   A = banded Gaussian weights A[m][j] = w[j-m] (registers, hoisted __expf)
//   B = input rows              B[j][n] = src[a0 + j - R][x0 + n]
//   Layout (CDNA5 ISA): A 16x4 f32: lanes0-15 K0/K1, lanes16-31
//   K2/K3; B mirrors the half-lane K split with N = lane&15; C/D: VGPR i ->
//   M=i (lo half) / M=8+i (hi half), N = lane&15  -> x-coalesced loads+stores.
// All geometry is compile-time (strides are powers of two -> shift address
// math); tile id forced wave-uniform via readfirstlane -> scalar base.
// Boundary rows: clamp + cndmask, no EXEC divergence (WMMA needs EXEC=all 1s).
// Each wave processes 4 x-subtiles (64 consecutive x), reusing the weights.
// ---------------------------------------------------------------------------
template <int K, int R, int SA, int SO, int NTA, int NOTHER, int AXLEN>
__global__ __launch_bounds__(256)
void blur_axis_wmma_kernel(const float* __restrict__ src, float* __restrict__ dst,
                           float invSigma, float invNorm) {
  constexpr int NCHUNK = (16 + K - 1 + 3) / 4;    // K=13 -> 7, K=7 -> 6
  const int lane = threadIdx.x & 31;
  const int tid =
      __builtin_amdgcn_readfirstlane(blockIdx.x * 8 + (threadIdx.x >> 5));

  // decomposition: [b][other][tai][txg] ; 4 x-groups of 64
  const int txg   = tid & 3;
  int rest        = tid >> 2;
  const int tai   = rest % NTA;
  rest           /= NTA;
  const int other = rest % NOTHER;
  const int b     = rest / NOTHER;

  const unsigned base =
      (unsigned)b * VOLB + (unsigned)other * SO + (unsigned)txg * 64u;
  const int a0 = tai * 16;

  const int m   = lane & 15;              // A row / B-D column index
  const int hi2 = (lane >> 4) << 1;       // K-pair select per lane half
  const int rowsel = a0 + hi2 - R;        // B row for chunk c: rowsel + 4c

  // A operand: banded Gaussian weights, hoisted out of the compute loop
  v2f aw[NCHUNK];
#pragma unroll
  for (int c = 0; c < NCHUNK; ++c) {
    const int k0 = 4 * c + hi2 - m;
    const int k1 = k0 + 1;
    const float t0 = (float)(k0 - R) * invSigma;
    const float t1 = (float)(k1 - R) * invSigma;
    aw[c].x = ((unsigned)k0 < (unsigned)K) ? __expf(-0.5f * t0 * t0) * invNorm : 0.f;
    aw[c].y = ((unsigned)k1 < (unsigned)K) ? __expf(-0.5f * t1 * t1) * invNorm : 0.f;
  }

  const int rowStoreBase = a0 + ((lane >> 4) << 3);

#pragma unroll
  for (int xt = 0; xt < 4; ++xt) {
    const unsigned col = base + (unsigned)(xt * 16) + (unsigned)m;

    // load phase: all B rows for this subtile (clamped, branch-free)
    v2f bv[NCHUNK];
#pragma unroll
    for (int c = 0; c < NCHUNK; ++c) {
      const int p0 = rowsel + 4 * c;
      const int p1 = p0 + 1;
      const int q0 = iclamp(p0, 0, AXLEN - 1);
      const int q1 = iclamp(p1, 0, AXLEN - 1);
      const float f0 = src[col + (unsigned)q0 * (unsigned)SA];
      const float f1 = src[col + (unsigned)q1 * (unsigned)SA];
      bv[c].x = ((unsigned)p0 < (unsigned)AXLEN) ? f0 : 0.f;
      bv[c].y = ((unsigned)p1 < (unsigned)AXLEN) ? f1 : 0.f;
    }

    // compute phase: WMMA accumulation chain
    v8f acc = {0.f, 0.f, 0.f, 0.f, 0.f, 0.f, 0.f, 0.f};
#pragma unroll
    for (int c = 0; c < NCHUNK; ++c)
      acc = __builtin_amdgcn_wmma_f32_16x16x4_f32(false, aw[c], false, bv[c],
                                                  (short)0, acc, false, false);

    // store D: 8 rows, constant stride -> shared address + immediate offsets
    const unsigned outBase = col + (unsigned)rowStoreBase * (unsigned)SA;
#pragma unroll
    for (int vi = 0; vi < 8; ++vi)
      dst[outBase + (unsigned)(vi * SA)] = acc[vi];
  }
}

// ---------------------------------------------------------------------------
// Pass 4: Hessian (19-pt stencil, x sigma^2) + eigenvalue approx + surfaceness,
// plus per-volume max via LDS block reduction + one atomicMax per block.
// Grid: (3072, batch) x 256 threads x 16 voxels = VOLB exactly.
// ---------------------------------------------------------------------------
__global__ __launch_bounds__(256)
void hessian_surf_kernel(const float* __restrict__ sm, float* __restrict__ surf,
                         unsigned* __restrict__ mx, int slot, float s2) {
  const int b = blockIdx.y;
  const float* smb = sm + (long long)b * VOLB;
  float* sfb = surf + (long long)b * VOLB;

  float lmax = 0.f;
#pragma unroll 1
  for (int i = 0; i < 16; ++i) {
    const int idx = blockIdx.x * 256 + threadIdx.x + i * (3072 * 256);
    const int x = idx & 255, y = (idx >> 8) & 255, z = idx >> 16;

    auto S = [&](int dz, int dy, int dx) -> float {
      const int zz = z + dz, yy = y + dy, xx = x + dx;
      const int zc = iclamp(zz, 0, DIMD - 1);
      const int yc = iclamp(yy, 0, DIMH - 1);
      const int xc = iclamp(xx, 0, DIMW - 1);
      const float f = smb[(zc << 16) + (yc << 8) + xc];
      const bool in = ((unsigned)zz < (unsigned)DIMD) &
                      ((unsigned)yy < (unsigned)DIMH) &
                      ((unsigned)xx < (unsigned)DIMW);
      return in ? f : 0.f;                 // 'SAME' zero padding
    };

    const float c0 = S(0, 0, 0);
    float fxx = (S(1, 0, 0) + S(-1, 0, 0) - 2.f * c0) * s2;   // dim0 (D)
    float fyy = (S(0, 1, 0) + S(0, -1, 0) - 2.f * c0) * s2;   // dim1 (H)
    float fzz = (S(0, 0, 1) + S(0, 0, -1) - 2.f * c0) * s2;   // dim2 (W)
    float fxy = 0.25f * (S(1, 1, 0) + S(-1, -1, 0) - S(1, -1, 0) - S(-1, 1, 0)) * s2;
    float fxz = 0.25f * (S(1, 0, 1) + S(-1, 0, -1) - S(1, 0, -1) - S(-1, 0, 1)) * s2;
    float fyz = 0.25f * (S(0, 1, 1) + S(0, -1, -1) - S(0, 1, -1) - S(0, -1, 1)) * s2;

    // eigenvalue approximation (mirrors reference exactly)
    float v1 = fxx + fxy + fxz, v2 = fxy + fyy + fyz, v3 = fxz + fyz + fzz;
    float inv = 1.f / sqrtf(v1 * v1 + v2 * v2 + v3 * v3 + EPSF);
    v1 *= inv; v2 *= inv; v3 *= inv;
    float Hv1 = fxx * v1 + fxy * v2 + fxz * v3;
    float Hv2 = fxy * v1 + fyy * v2 + fyz * v3;
    float Hv3 = fxz * v1 + fyz * v2 + fzz * v3;
    float l3 = Hv1 * v1 + Hv2 * v2 + Hv3 * v3;
    float trace = fxx + fyy + fzz;
    float det = fxx * (fyy * fzz - fyz * fyz) - fxy * (fxy * fzz - fxz * fyz) +
                fxz * (fxy * fyz - fxz * fyy);
    float s = trace - l3;
    float p = det / (l3 + EPSF);
    float disc = fmaxf(s * s - 4.f * p, 0.f);
    float sd = sqrtf(disc + EPSF);
    float e0 = 0.5f * (s - sd), e1 = 0.5f * (s + sd), e2 = l3;
    // sort by |lambda| ascending
    if (fabsf(e0) > fabsf(e1)) { float t = e0; e0 = e1; e1 = t; }
    if (fabsf(e1) > fabsf(e2)) { float t = e1; e1 = e2; e2 = t; }
    if (fabsf(e0) > fabsf(e1)) { float t = e0; e0 = e1; e1 = t; }

    float sv = __expf(-(e0 * e0 + e1 * e1) / (0.25f * (e2 * e2 + EPSF))) * fabsf(e2);
    sfb[idx] = sv;
    lmax = fmaxf(lmax, sv);
  }

  // block max reduction in LDS, single atomic per block
  __shared__ float red[256];
  red[threadIdx.x] = lmax;
  __syncthreads();
  for (int st = 128; st > 0; st >>= 1) {
    if (threadIdx.x < st)
      red[threadIdx.x] = fmaxf(red[threadIdx.x], red[threadIdx.x + st]);
    __syncthreads();
  }
  if (threadIdx.x == 0)
    atomicMax(&mx[2 * b + slot], __float_as_uint(red[0]));  // surf >= 0: uint order == float order
}

// ---------------------------------------------------------------------------
// Pass 5: out = filt * max( surf1/(mx1+eps), surf2/(mx2+eps) ) with mx>0 guard
// ---------------------------------------------------------------------------
__global__ __launch_bounds__(256)
void finalize_kernel(const float* __restrict__ v, const float* __restrict__ s1,
                     const float* __restrict__ s2, const unsigned* __restrict__ mx,
                     float* __restrict__ out) {
  const int idx = blockIdx.x * 256 + threadIdx.x;
  const int b = (idx >= VOLB) ? 1 : 0;
  const float m1 = __uint_as_float(mx[2 * b + 0]);
  const float m2 = __uint_as_float(mx[2 * b + 1]);
  const float a = (m1 > 0.f) ? s1[idx] / (m1 + EPSF) : 0.f;
  const float c = (m2 > 0.f) ? s2[idx] / (m2 + EPSF) : 0.f;
  const float filt = (v[idx] >= 0.75f) ? 1.f : 0.f;
  out[idx] = filt * fmaxf(a, c);
}

// ---------------------------------------------------------------------------
// host side
// ---------------------------------------------------------------------------
template <int K>
static void fill_gauss(WgtArr<K>& wg, double sigma, float* invNorm) {
  const int r = (K - 1) / 2;
  double s = 0.0;
  double tmp[K];
  for (int i = 0; i < K; ++i) {
    double x = (double)(i - r);
    tmp[i] = exp(-0.5 * x * x / (sigma * sigma));
    s += tmp[i];
  }
  for (int i = 0; i < K; ++i) wg.w[i] = (float)(tmp[i] / s);
  *invNorm = (float)(1.0 / s);
}

extern "C" void kernel_launch(void* const* d_in, const int* in_sizes, int n_in,
                              void* d_out, int out_size, void* d_ws, size_t ws_size,
                              hipStream_t stream) {
  const float* v = (const float*)d_in[0];
  float* out = (float*)d_out;

  // workspace layout: 3 full-tensor f32 buffers + 4 max words
  float* t0 = (float*)d_ws;
  float* t1 = t0 + (size_t)NTOT;
  float* t2 = t1 + (size_t)NTOT;
  unsigned* mx = (unsigned*)(t2 + (size_t)NTOT);

  WgtArr<13> w13; float inv13;
  WgtArr<7>  w7;  float inv7;
  fill_gauss<13>(w13, 2.0, &inv13);
  fill_gauss<7>(w7, 1.0, &inv7);

  const dim3 blk(256);
  const int elemBlocks = NTOT / 256;   // 98304
  const int tileBlocks = 3072;         // (2*other*NTA*4 xgroups)/8 waves = 3072 both axes

  init_max_kernel<<<1, 32, 0, stream>>>(mx);

  // ---- sigma = 2 (13 taps, r=6): v ->t0 ->t1 ->t0 ; surf -> t1, max slot 1
  blur_x_thresh_kernel<13, 6><<<elemBlocks, blk, 0, stream>>>(v, t0, w13);
  blur_axis_wmma_kernel<13, 6, /*SA=*/DIMW, /*SO=*/DIMH * DIMW,
                        /*NTA=*/DIMH / 16, /*NOTHER=*/DIMD, /*AXLEN=*/DIMH>
      <<<tileBlocks, blk, 0, stream>>>(t0, t1, 0.5f, inv13);
  blur_axis_wmma_kernel<13, 6, /*SA=*/DIMH * DIMW, /*SO=*/DIMW,
                        /*NTA=*/DIMD / 16, /*NOTHER=*/DIMH, /*AXLEN=*/DIMD>
      <<<tileBlocks, blk, 0, stream>>>(t1, t0, 0.5f, inv13);
  hessian_surf_kernel<<<dim3(3072, 2), blk, 0, stream>>>(t0, t1, mx, 1, 4.0f);

  // ---- sigma = 1 (7 taps, r=3): v ->t0 ->t2 ->t0 ; surf -> t2, max slot 0
  blur_x_thresh_kernel<7, 3><<<elemBlocks, blk, 0, stream>>>(v, t0, w7);
  blur_axis_wmma_kernel<7, 3, DIMW, DIMH * DIMW, DIMH / 16, DIMD, DIMH>
      <<<tileBlocks, blk, 0, stream>>>(t0, t2, 1.0f, inv7);
  blur_axis_wmma_kernel<7, 3, DIMH * DIMW, DIMW, DIMD / 16, DIMH, DIMD>
      <<<tileBlocks, blk, 0, stream>>>(t2, t0, 1.0f, inv7);
  hessian_surf_kernel<<<dim3(3072, 2), blk, 0, stream>>>(t0, t2, mx, 0, 1.0f);

  // ---- combine
  finalize_kernel<<<elemBlocks, blk, 0, stream>>>(v, t2, t1, mx, out);
}